// MoERouter_57595511439903
// MI455X (gfx1250) — compile-verified
//
#include <hip/hip_runtime.h>
#include <hip/hip_bf16.h>

typedef __attribute__((ext_vector_type(2))) float v2f;
typedef __attribute__((ext_vector_type(4))) float v4f;
typedef __attribute__((ext_vector_type(8))) float v8f;

// ---------------------------------------------------------------------------
// Kernel 1: router GEMV via V_WMMA_F32_16X16X4_F32.
//   A (16x4 f32): row 0 = W[k..k+3] (lanes 0 and 16 carry it), rows 1..15 = 0.
//   B (4x16 f32): B[k][n] = H[t0+n][kb+k]  (16 token rows, transposed chunk).
//   D[0][n] accumulates score of token t0+n -> c[0] on lanes 0..15.
// Memory-bound: per lane one b128 load of H per K-step; lanes L and L+16 load
// the same 16B (cache-merged), 16 unique rows x 16B per instruction, rows are
// streamed contiguously over the K loop.
// ---------------------------------------------------------------------------
__global__ void __launch_bounds__(256)
router_gemv_wmma(const float* __restrict__ H,
                 const unsigned char* __restrict__ amask,
                 const float* __restrict__ U,
                 const float* __restrict__ W,
                 const float* __restrict__ bias,
                 float* __restrict__ ws_scores,
                 unsigned int* __restrict__ ws_keys,
                 int D, int N)
{
    const int lane = threadIdx.x & 31;
    const int wave = threadIdx.x >> 5;
    const int t0   = (blockIdx.x * 8 + wave) * 16;   // 8 waves/block, 16 tokens/wave
    if (t0 >= N) return;

    int row = t0 + (lane & 15);
    if (row >= N) row = N - 1;                        // safety clamp (shapes divide evenly)
    const float* __restrict__ hrow = H + (size_t)row * (size_t)D;

    const bool hiHalf = (lane >= 16);
    const bool isL0   = (lane == 0);
    const bool isL16  = (lane == 16);

    v8f c = {0.f, 0.f, 0.f, 0.f, 0.f, 0.f, 0.f, 0.f};

    for (int kb = 0; kb < D; kb += 16) {
#pragma unroll
        for (int j = 0; j < 4; ++j) {
            const int k = kb + j * 4;
            // Per-lane 16B of hidden state (lanes L and L+16 -> same address).
            v4f h = *(const v4f*)(hrow + k);
            // Uniform address -> scalar load of 4 weights.
            v4f w = *(const v4f*)(W + k);

            // B-matrix vgprs: vgpr0 = rows K=0 (lanes 0-15) / K=2 (lanes 16-31)
            //                 vgpr1 = rows K=1 / K=3
            v2f Bm;
            Bm.x = hiHalf ? h.z : h.x;
            Bm.y = hiHalf ? h.w : h.y;
            // A-matrix vgprs: lane0 holds M=0,K=0/1 ; lane16 holds M=0,K=2/3 ; rest 0
            v2f Am;
            Am.x = isL0 ? w.x : (isL16 ? w.z : 0.0f);
            Am.y = isL0 ? w.y : (isL16 ? w.w : 0.0f);

            c = __builtin_amdgcn_wmma_f32_16x16x4_f32(
                    /*neg_a=*/false, Am, /*neg_b=*/false, Bm,
                    /*c_mod=*/(short)0, c, /*reuse_a=*/false, /*reuse_b=*/false);
        }
    }

    // Scores for tokens t0..t0+15 live in c[0] on lanes 0..15.
    if (lane < 16) {
        const int t = t0 + lane;
        if (t < N) {
            float score = c[0] + bias[0];
            const bool act = (amask[t] != 0);
            score = act ? score : -1e9f;

            // gumbel noise (TEMPERATURE = 1)
            float u = U[t];
            u = fminf(fmaxf(u, 1e-6f), 1.0f - 1e-6f);
            float g = -logf(-logf(u) + 1e-6f);
            g = fminf(fmaxf(g, -10.0f), 10.0f);
            const float noisy = score + g;

            // order-preserving float -> uint key (ascending)
            unsigned int ub = __float_as_uint(noisy);
            unsigned int key = (ub & 0x80000000u) ? ~ub : (ub | 0x80000000u);

            ws_scores[t] = score;
            ws_keys[t]   = key;
        }
    }
}

// ---------------------------------------------------------------------------
// Kernel 2 (single workgroup, deterministic):
//   (a) aux-loss reductions over scores (LDS tree reduce)
//   (b) exact k-th-largest key via 4-pass 8-bit radix select (integer LDS
//       atomics -> deterministic)
//   (c) mask write: key >= threshold  (& active_mask)
// ---------------------------------------------------------------------------
__global__ void __launch_bounds__(1024)
router_select_finalize(const float* __restrict__ ws_scores,
                       const unsigned int* __restrict__ ws_keys,
                       const unsigned char* __restrict__ amask,
                       float* __restrict__ out_mask,
                       float* __restrict__ out_aux,
                       int N, int K)
{
    __shared__ float s_p[1024];
    __shared__ float s_z[1024];
    __shared__ float s_e[1024];
    __shared__ unsigned int hist[256];
    __shared__ unsigned int s_prefix;
    __shared__ unsigned int s_remaining;

    const int tid = threadIdx.x;

    // ---- (a) aux loss reductions ----
    float sp = 0.f, sz = 0.f, se = 0.f;
    for (int i = tid; i < N; i += 1024) {
        const float s = ws_scores[i];
        const float p = 1.0f / (1.0f + expf(-s));
        sp += p;
        sz += s * s;
        float pc = fminf(fmaxf(p, 1e-4f), 1.0f - 1e-4f);
        se += pc * logf(pc) + (1.0f - pc) * logf(1.0f - pc);   // = -entropy term
    }
    s_p[tid] = sp; s_z[tid] = sz; s_e[tid] = se;
    __syncthreads();
    for (int off = 512; off > 0; off >>= 1) {
        if (tid < off) {
            s_p[tid] += s_p[tid + off];
            s_z[tid] += s_z[tid + off];
            s_e[tid] += s_e[tid + off];
        }
        __syncthreads();
    }
    if (tid == 0) {
        const float invN   = 1.0f / (float)N;
        const float mean_p = s_p[0] * invN;
        const float lb     = (mean_p - 0.55f) * (mean_p - 0.55f);
        const float zl     = s_z[0] * invN;
        const float entl   = s_e[0] * invN;   // entropy_loss = -entropy = mean(se)
        out_aux[0] = 0.01f * lb + 1e-4f * zl + 1e-3f * entl;
        s_prefix = 0u;
        s_remaining = (unsigned int)K;
    }
    __syncthreads();

    // ---- (b) radix select: k-th largest key ----
    for (int pass = 3; pass >= 0; --pass) {
        const int shift = pass * 8;
        if (tid < 256) hist[tid] = 0u;
        __syncthreads();

        const unsigned int prefix = s_prefix;
        const unsigned int pmask  = (pass == 3) ? 0u : (0xFFFFFFFFu << (shift + 8));
        for (int i = tid; i < N; i += 1024) {
            const unsigned int key = ws_keys[i];
            if ((key & pmask) == prefix)
                atomicAdd(&hist[(key >> shift) & 0xFFu], 1u);
        }
        __syncthreads();

        if (tid == 0) {
            unsigned int rem = s_remaining;
            unsigned int cum = 0u;
            int sel = 0;
            for (int b = 255; b >= 0; --b) {
                const unsigned int h = hist[b];
                if (cum + h >= rem) { sel = b; rem -= cum; break; }
                cum += h;
            }
            s_prefix    = prefix | ((unsigned int)sel << shift);
            s_remaining = rem;
        }
        __syncthreads();
    }

    // ---- (c) mask write ----
    const unsigned int thr = s_prefix;
    for (int i = tid; i < N; i += 1024) {
        const bool sel = (ws_keys[i] >= thr) && (amask[i] != 0);
        out_mask[i] = sel ? 1.0f : 0.0f;
    }
}

// ---------------------------------------------------------------------------
// Launch wrapper
// inputs: 0 hidden_states f32[B,S,D], 1 active_mask bool[B,S],
//         2 uniform f32[B,S], 3 W f32[D], 4 b f32[1]
// output: [N] mask as 0/1 floats, then aux_loss scalar at index N.
// ---------------------------------------------------------------------------
extern "C" void kernel_launch(void* const* d_in, const int* in_sizes, int n_in,
                              void* d_out, int out_size, void* d_ws, size_t ws_size,
                              hipStream_t stream)
{
    const float*         H     = (const float*)d_in[0];
    const unsigned char* amask = (const unsigned char*)d_in[1];
    const float*         U     = (const float*)d_in[2];
    const float*         W     = (const float*)d_in[3];
    const float*         bias  = (const float*)d_in[4];

    const int N = in_sizes[1];   // B*S tokens
    const int D = in_sizes[3];   // hidden dim

    float*        out       = (float*)d_out;
    float*        ws_scores = (float*)d_ws;
    unsigned int* ws_keys   = (unsigned int*)((char*)d_ws + (size_t)N * sizeof(float));

    int K = (int)(0.55 * (double)N + 0.5);
    if (K < 1) K = 1;
    if (K > N) K = N;

    const int tokensPerBlock = 128;  // 8 waves x 16 tokens
    dim3 grid((N + tokensPerBlock - 1) / tokensPerBlock);
    router_gemv_wmma<<<grid, 256, 0, stream>>>(H, amask, U, W, bias,
                                               ws_scores, ws_keys, D, N);

    router_select_finalize<<<1, 1024, 0, stream>>>(ws_scores, ws_keys, amask,
                                                   out, out + N, N, K);
}